// EngramGating_29360396435933
// MI455X (gfx1250) — compile-verified
//
#include <hip/hip_runtime.h>
#include <cstdint>

typedef float v2f  __attribute__((ext_vector_type(2)));
typedef float v8f  __attribute__((ext_vector_type(8)));
typedef unsigned int u32x4 __attribute__((ext_vector_type(4)));
typedef int   i32x4 __attribute__((ext_vector_type(4)));
typedef int   i32x8 __attribute__((ext_vector_type(8)));

#define EPS_F        1.1920928955078125e-07f   /* FLT_EPSILON (torch RMSNorm default) */
#define INV_SQRT_DIM 0.17677669529663687f      /* 1/sqrt(32) */

// Padded LDS layout: each 32-dword weight row gets +2 dwords of padding so the
// per-lane fragment reads (stride 34 dwords) are LDS-bank-conflict free.
#define ROW_STRIDE 34
#define WK_ROWS    128
#define WV_ROWS    32
#define WV_OFF     (WK_ROWS * ROW_STRIDE)              /* 4352 floats            */
#define SMEM_FLOATS (WV_OFF + WV_ROWS * ROW_STRIDE)    /* 5440 floats = 21.76 KB */

#if __has_include(<hip/amd_detail/amd_gfx1250_TDM.h>)
#define ATHENA_TDM_ARITY6 1
#endif

// ---- f32 WMMA: D(16x16) = A(16x4) * B(4x16) + C, chained over K ----
__device__ __forceinline__ v8f wmma_f32_k4(v2f a, v2f b, v8f c) {
  return __builtin_amdgcn_wmma_f32_16x16x4_f32(false, a, false, b, (short)0, c,
                                               false, false);
}

// sum within each 16-lane half of a wave32 (halves stay independent)
__device__ __forceinline__ float red16(float x) {
  x += __shfl_xor(x, 1, 32);
  x += __shfl_xor(x, 2, 32);
  x += __shfl_xor(x, 4, 32);
  x += __shfl_xor(x, 8, 32);
  return x;
}

#if defined(__AMDGCN__) && __has_builtin(__builtin_amdgcn_tensor_load_to_lds)
// 2-D TDM transfer: nrows x 32 f32 rows from global -> LDS at byte lds_off,
// with TDM-inserted padding of 2 dwords after every 32 dwords stored, so the
// LDS image has a 34-dword row stride (bank-conflict-free fragment reads).
__device__ __forceinline__ void tdm_load_rows_padded(const float* gptr,
                                                     unsigned lds_off,
                                                     unsigned nrows) {
  unsigned long long ga = (unsigned long long)(uintptr_t)gptr;
  u32x4 g0;
  g0.x = 1u;                                                 // count=1 (valid D#)
  g0.y = lds_off;                                            // lds_addr (bytes)
  g0.z = (unsigned)(ga & 0xFFFFFFFFull);                     // global_addr[31:0]
  g0.w = (unsigned)((ga >> 32) & 0x1FFFFFFull) | (2u << 30); // addr[56:32]|type=2
  i32x8 g1;
  // data_size=4B | pad_enable | pad_interval=4 (16 x 8B = 32 dwords) |
  // pad_amount=1 (2 dwords)
  g1[0] = (int)((2u << 16) | (1u << 20) | (4u << 22) | (1u << 25));
  g1[1] = (int)(32u << 16);             // tensor_dim0 lo16 = 32 (bits 63:48)
  g1[2] = (int)(nrows << 16);           // dim0 hi16=0 | tensor_dim1 lo16 = nrows
  g1[3] = (int)(32u << 16);             // dim1 hi16=0 | tile_dim0 = 32
  g1[4] = (int)nrows;                   // tile_dim1 = nrows, tile_dim2 = 0
  g1[5] = 32;                           // tensor_dim0_stride lo32 = 32
  g1[6] = 0;
  g1[7] = 0;
  i32x4 z4 = {0, 0, 0, 0};
#ifdef ATHENA_TDM_ARITY6
  i32x8 z8 = {0, 0, 0, 0, 0, 0, 0, 0};
  __builtin_amdgcn_tensor_load_to_lds(g0, g1, z4, z4, z8, 0);
#else
  __builtin_amdgcn_tensor_load_to_lds(g0, g1, z4, z4, 0);
#endif
}
#endif

__global__ __launch_bounds__(256, 5) void engram_gating_kernel(
    const float* __restrict__ emb, const float* __restrict__ hid,
    const float* __restrict__ Wv,  const float* __restrict__ bv,
    const float* __restrict__ Wk,  const float* __restrict__ bk,
    const float* __restrict__ g1,  const float* __restrict__ g2,
    float* __restrict__ out, int tiles) {
  // [0, 4352): Wk (128 rows, 34-dword stride), [4352, 5440): Wv (32 rows)
  __shared__ float smem[SMEM_FLOATS];

  const int tid = threadIdx.x;

  // ---- stage weights into LDS via the Tensor Data Mover (wave 0 only) ----
#if defined(__AMDGCN__) && __has_builtin(__builtin_amdgcn_tensor_load_to_lds)
  if (tid < 32) {
    tdm_load_rows_padded(Wk, 0u, WK_ROWS);
    tdm_load_rows_padded(Wv, WV_OFF * 4u, WV_ROWS);
    __builtin_amdgcn_s_wait_tensorcnt(0);
  }
#else
  for (int i = tid; i < 4096; i += 256)
    smem[(i >> 5) * ROW_STRIDE + (i & 31)] = Wk[i];
  for (int i = tid; i < 1024; i += 256)
    smem[WV_OFF + (i >> 5) * ROW_STRIDE + (i & 31)] = Wv[i];
#endif
  // The TDM writes LDS through the descriptor, invisibly to alias analysis.
  // Escape the smem pointer through an asm memory clobber so the compiler
  // treats the LDS contents as externally defined (prevents undef-folding).
  asm volatile("" : : "r"(&smem[0]) : "memory");
  __syncthreads();

  const int lane = tid & 31;
  const int wave = tid >> 5;
  const int row  = lane & 15;  // N index / token row
  const int half = lane >> 4;  // K-half for A/B frags, M-half for C/D

  // value bias (broadcast along M in the C/D layout)
  const float bvf0 = bv[row];
  const float bvf1 = bv[16 + row];

  const int wave_id = blockIdx.x * 8 + wave;
  const int n_waves = gridDim.x * 8;

  for (int tile = wave_id; tile < tiles; tile += n_waves) {
    const int t0 = tile * 16;

    // prefetch next tile's streams into the cache hierarchy
    const int nt = tile + n_waves;
    if (nt < tiles) {
      const int p0 = nt * 16;
      __builtin_prefetch(&emb[p0 * 32 + lane * 16], 0, 0);
      __builtin_prefetch(&hid[p0 * 128 + lane * 32], 0, 0);
      __builtin_prefetch(&hid[p0 * 128 + 4096 + lane * 32], 0, 0);
    }

    // A-fragments: emb tile [16 tokens x 32 dims]
    v2f a[8];
#pragma unroll
    for (int j = 0; j < 8; ++j)
      a[j] = *(const v2f*)&emb[(t0 + row) * 32 + j * 4 + half * 2];

    // value = emb @ Wv^T + bv  (two 16-wide N tiles); Wv B-fragments are
    // transient, re-read from the padded (conflict-free) LDS image.
    v8f val0, val1;
#pragma unroll
    for (int r = 0; r < 8; ++r) { val0[r] = bvf0; val1[r] = bvf1; }
#pragma unroll
    for (int j = 0; j < 8; ++j) {
      v2f w0 = *(const v2f*)&smem[WV_OFF + row * ROW_STRIDE + j * 4 + half * 2];
      v2f w1 = *(const v2f*)&smem[WV_OFF + (16 + row) * ROW_STRIDE + j * 4 + half * 2];
      val0 = wmma_f32_k4(a[j], w0, val0);
      val1 = wmma_f32_k4(a[j], w1, val1);
    }

    // Heads processed strictly sequentially to bound register pressure:
    // only one head's Wk fragments / key accumulators / queries live at once.
#pragma unroll 1
    for (int h = 0; h < 4; ++h) {
      // per-head bias & RMSNorm gains (tiny, L0-resident loads)
      const float bk0 = bk[h * 32 + row],      bk1 = bk[h * 32 + 16 + row];
      const float g10 = g1[h * 32 + row],      g11 = g1[h * 32 + 16 + row];
      const float g20 = g2[h * 32 + row],      g21 = g2[h * 32 + 16 + row];

      // Wk B-fragments for this head (padded LDS, conflict-free)
      v2f wkf0[8], wkf1[8];
#pragma unroll
      for (int j = 0; j < 8; ++j) {
        wkf0[j] = *(const v2f*)&smem[(h * 32 + row) * ROW_STRIDE + j * 4 + half * 2];
        wkf1[j] = *(const v2f*)&smem[(h * 32 + 16 + row) * ROW_STRIDE + j * 4 + half * 2];
      }

      // key = emb @ Wk[h]^T + bk[h]
      v8f key0, key1;
#pragma unroll
      for (int r = 0; r < 8; ++r) { key0[r] = bk0; key1[r] = bk1; }
#pragma unroll
      for (int j = 0; j < 8; ++j) {
        key0 = wmma_f32_k4(a[j], wkf0[j], key0);
        key1 = wmma_f32_k4(a[j], wkf1[j], key1);
      }

      // queries in the same (token M = r + 8*half, k = n*16 + row) layout
      float q0[8], q1[8];
#pragma unroll
      for (int r = 0; r < 8; ++r) {
        const int t = t0 + r + half * 8;
        q0[r] = hid[(t * 4 + h) * 32 + row];
        q1[r] = hid[(t * 4 + h) * 32 + 16 + row];
      }

#pragma unroll
      for (int r = 0; r < 8; ++r) {
        float msk = red16(key0[r] * key0[r] + key1[r] * key1[r]);
        float msq = red16(q0[r] * q0[r] + q1[r] * q1[r]);
        float rk = rsqrtf(msk * 0.03125f + EPS_F);
        float rq = rsqrtf(msq * 0.03125f + EPS_F);
        float nk0 = key0[r] * rk * g10, nk1 = key1[r] * rk * g11;
        float nq0 = q0[r] * rq * g20,   nq1 = q1[r] * rq * g21;
        float g = red16(nk0 * nq0 + nk1 * nq1) * INV_SQRT_DIM;
        float sg = (g > 0.f) ? 1.f : ((g < 0.f) ? -1.f : 0.f);
        float s = sqrtf(fmaxf(fabsf(g), 1e-6f)) * sg;
        float gate = 1.f / (1.f + __expf(-s));
        const int t = t0 + r + half * 8;
        out[(t * 4 + h) * 32 + row]      = gate * val0[r];
        out[(t * 4 + h) * 32 + 16 + row] = gate * val1[r];
      }
    }
  }
}

extern "C" void kernel_launch(void* const* d_in, const int* in_sizes, int n_in,
                              void* d_out, int out_size, void* d_ws, size_t ws_size,
                              hipStream_t stream) {
  const float* emb = (const float*)d_in[0];
  const float* hid = (const float*)d_in[1];
  const float* Wv  = (const float*)d_in[2];
  const float* bv  = (const float*)d_in[3];
  const float* Wk  = (const float*)d_in[4];
  const float* bk  = (const float*)d_in[5];
  const float* g1  = (const float*)d_in[6];
  const float* g2  = (const float*)d_in[7];
  float* out = (float*)d_out;

  const int T = in_sizes[0] / 32;   // B*S tokens
  const int tiles = T / 16;         // 16-token tiles, one per wave iteration
  int blocks = (tiles + 31) / 32;   // 8 waves/block, ~4 tiles per wave
  if (blocks > 2048) blocks = 2048;
  if (blocks < 1) blocks = 1;

  engram_gating_kernel<<<blocks, 256, 0, stream>>>(emb, hid, Wv, bv, Wk, bk,
                                                   g1, g2, out, tiles);
}